// MultiHeadSelfAttentionSDPA_30288109371668
// MI455X (gfx1250) — compile-verified
//
#include <hip/hip_runtime.h>

typedef float v2f __attribute__((ext_vector_type(2)));
typedef float v8f __attribute__((ext_vector_type(8)));

#define D_MODEL 2048
#define N_HEADS 16
#define D_K     128
#define SEQ     2048

// ---------------------------------------------------------------------------
// CDNA5 async global->LDS copies (ASYNCcnt-tracked DMA), with sync fallback.
// Builtin signatures (from hipcc diagnostics):
//   b32: (as1 int*, as3 int*, imm int offset, imm int cpol)
//   b64: (as1 v2i*, as3 v2i*, imm int offset, imm int cpol)   v2i = vector_size(8) int
// ---------------------------------------------------------------------------
#if __has_builtin(__builtin_amdgcn_global_load_async_to_lds_b64) && \
    __has_builtin(__builtin_amdgcn_global_load_async_to_lds_b32) && \
    __has_builtin(__builtin_amdgcn_s_wait_asynccnt)
#define USE_ASYNC_LDS 1
#else
#define USE_ASYNC_LDS 0
#endif

typedef int gv2i __attribute__((vector_size(8)));
typedef __attribute__((address_space(1))) gv2i* as1_v2i_p;
typedef __attribute__((address_space(3))) gv2i* as3_v2i_p;
typedef __attribute__((address_space(1))) int*  as1_i32_p;
typedef __attribute__((address_space(3))) int*  as3_i32_p;

__device__ __forceinline__ void cp_lds_b64(float* l, const float* g) {
#if USE_ASYNC_LDS
    __builtin_amdgcn_global_load_async_to_lds_b64(
        (as1_v2i_p)(uintptr_t)g,
        (as3_v2i_p)(unsigned)(uintptr_t)l, 0, 0);
#else
    *(v2f*)l = *(const v2f*)g;
#endif
}
__device__ __forceinline__ void cp_lds_b32(float* l, const float* g) {
#if USE_ASYNC_LDS
    __builtin_amdgcn_global_load_async_to_lds_b32(
        (as1_i32_p)(uintptr_t)g,
        (as3_i32_p)(unsigned)(uintptr_t)l, 0, 0);
#else
    *l = *g;
#endif
}
#if USE_ASYNC_LDS
#define WAIT_ASYNC(n) __builtin_amdgcn_s_wait_asynccnt(n)
#else
#define WAIT_ASYNC(n) ((void)0)
#endif

// ---- WMMA helper: full-precision fp32 matrix op (16x16x4) -------------------
__device__ __forceinline__ v8f wmma_f32(v2f a, v2f b, v8f c) {
    return __builtin_amdgcn_wmma_f32_16x16x4_f32(
        false, a, false, b, (short)0, c, false, false);
}

// =============================================================================
// GEMM: C[M,N] = A[M,K] @ W[N,K]^T
// 256 threads = 8 waves; 128x128 macro tile; K staged 16-wide, double-buffered
// via async-to-LDS. Wave grid 4x2: each wave 32 rows x 64 cols (2x4 acc tiles).
// =============================================================================
#define BM   128
#define BN   128
#define KT   16
#define LSTR 20   // 16 data + 4 pad: conflict-free (gcd(20,64)=4), 8B-aligned float2

__device__ __forceinline__ void stage_panels(float* AsB, float* WsB,
                                             const float* __restrict__ A,
                                             const float* __restrict__ W,
                                             size_t aoff, size_t woff,
                                             int K, int tid) {
    #pragma unroll
    for (int i = 0; i < 4; ++i) {
        int f   = tid + i * 256;   // float2 index 0..1023 (128 rows x 8)
        int row = f >> 3;
        int c2  = (f & 7) * 2;
        cp_lds_b64(&AsB[row * LSTR + c2], A + aoff + (size_t)row * K + c2);
        cp_lds_b64(&WsB[row * LSTR + c2], W + woff + (size_t)row * K + c2);
    }   // 8 async b64 per thread
}

__global__ __launch_bounds__(256)
void gemm_xwt(const float* __restrict__ A, const float* __restrict__ W,
              float* __restrict__ C, int M, int N, int K) {
    __shared__ float As[2][BM * LSTR];
    __shared__ float Ws[2][BN * LSTR];

    const int tid   = threadIdx.x;
    const int wave  = tid >> 5;
    const int lane  = tid & 31;
    const int half  = lane >> 4;
    const int r     = lane & 15;
    const int waveM = wave >> 1;   // 0..3 -> 32-row strip
    const int waveN = wave & 1;    // 0..1 -> 64-col strip

    const int mBase = blockIdx.y * BM;
    const int nBase = blockIdx.x * BN;

    v8f acc[2][4];
    #pragma unroll
    for (int mt = 0; mt < 2; ++mt)
        #pragma unroll
        for (int nt = 0; nt < 4; ++nt) acc[mt][nt] = {};

    const int NT = K / KT;
    stage_panels(As[0], Ws[0], A, W, (size_t)mBase * K, (size_t)nBase * K, K, tid);

    for (int kt = 0; kt < NT; ++kt) {
        const int buf = kt & 1;
        if (kt + 1 < NT) {
            size_t k0n = (size_t)(kt + 1) * KT;
            stage_panels(As[buf ^ 1], Ws[buf ^ 1], A, W,
                         (size_t)mBase * K + k0n, (size_t)nBase * K + k0n, K, tid);
            WAIT_ASYNC(8);    // next tile's 8 copies may remain in flight
        } else {
            WAIT_ASYNC(0);
        }
        __syncthreads();

        const float* AsB = As[buf];
        const float* WsB = Ws[buf];
        #pragma unroll
        for (int kk = 0; kk < 4; ++kk) {
            v2f af[2], bf[4];
            #pragma unroll
            for (int mt = 0; mt < 2; ++mt)
                af[mt] = *(const v2f*)(&AsB[(waveM * 32 + mt * 16 + r) * LSTR +
                                            kk * 4 + 2 * half]);
            #pragma unroll
            for (int nt = 0; nt < 4; ++nt)
                bf[nt] = *(const v2f*)(&WsB[(waveN * 64 + nt * 16 + r) * LSTR +
                                            kk * 4 + 2 * half]);
            #pragma unroll
            for (int mt = 0; mt < 2; ++mt)
                #pragma unroll
                for (int nt = 0; nt < 4; ++nt)
                    acc[mt][nt] = wmma_f32(af[mt], bf[nt], acc[mt][nt]);
        }
        __syncthreads();   // protect buf from being overwritten by stage kt+2
    }

    // C layout: VGPR p -> row p + 8*half, lane r -> col
    #pragma unroll
    for (int mt = 0; mt < 2; ++mt)
        #pragma unroll
        for (int nt = 0; nt < 4; ++nt)
            #pragma unroll
            for (int p = 0; p < 8; ++p) {
                int row = mBase + waveM * 32 + mt * 16 + p + 8 * half;
                int col = nBase + waveN * 64 + nt * 16 + r;
                C[(size_t)row * N + col] = acc[mt][nt][p];
            }
}

// =============================================================================
// Flash-attention with softcap. 4 waves/block, each wave a 16-row query tile;
// K/V tiles double-buffered in LDS via async copies.
// =============================================================================
#define KSTR 132   // K tile [key][d], 128+4
#define VSTR 20    // V tile transposed [d][key], 16+4
#define PSTR 20    // per-wave P staging

__device__ __forceinline__ void stage_kv(float* KsB, float* VtB,
                                         const float* __restrict__ Kb,
                                         const float* __restrict__ Vb,
                                         size_t gbase, int tid) {
    // K tile 16x128 as float2 (1024 / 128 thr = 8 each)
    #pragma unroll
    for (int i = 0; i < 8; ++i) {
        int f   = tid + i * 128;
        int row = f >> 6;            // 64 float2 per row
        int c2  = (f & 63) * 2;
        cp_lds_b64(&KsB[row * KSTR + c2], Kb + gbase + (size_t)row * D_MODEL + c2);
    }
    // V tile transposed, scalar scatter (2048 / 128 thr = 16 each)
    #pragma unroll
    for (int i = 0; i < 16; ++i) {
        int idx = tid + i * 128;
        int key = idx >> 7;
        int d   = idx & 127;
        cp_lds_b32(&VtB[d * VSTR + key], Vb + gbase + (size_t)key * D_MODEL + d);
    }   // 24 asyncs per thread total
}

__global__ __launch_bounds__(128)
void attn_softcap(const float* __restrict__ Q, const float* __restrict__ Kb,
                  const float* __restrict__ Vb, const float* __restrict__ bias,
                  float* __restrict__ Ctx, const int* __restrict__ capPtr) {
    __shared__ float Ks[2][16 * KSTR];
    __shared__ float Vt[2][D_K * VSTR];
    __shared__ float Pst[4][16 * PSTR];

    const int tid  = threadIdx.x;
    const int wave = tid >> 5;
    const int lane = tid & 31;
    const int half = lane >> 4;
    const int r    = lane & 15;

    const int bh = blockIdx.y;
    const int b  = bh >> 4;
    const int h  = bh & (N_HEADS - 1);
    const int qBase = blockIdx.x * 64 + wave * 16;

    const size_t headOff = (size_t)h * D_K;
    const float  scale   = 0.08838834764831845f;  // 1/sqrt(128)
    const float  cap     = (float)capPtr[0];
    const float  invCap  = 1.0f / cap;

    // Q fragments: 32 chunks of K=4 over d_k=128
    v2f qf[32];
    {
        const float* qrow = Q + (size_t)(b * SEQ + qBase + r) * D_MODEL + headOff;
        #pragma unroll
        for (int kc = 0; kc < 32; ++kc)
            qf[kc] = *(const v2f*)(qrow + kc * 4 + 2 * half);
    }

    v8f oacc[8];
    #pragma unroll
    for (int i = 0; i < 8; ++i) oacc[i] = {};
    float mrow[8], lrow[8];
    #pragma unroll
    for (int p = 0; p < 8; ++p) { mrow[p] = -3.0e38f; lrow[p] = 0.0f; }

    const int NT = SEQ / 16;
    const size_t bhBase = (size_t)(b * SEQ) * D_MODEL + headOff;
    stage_kv(Ks[0], Vt[0], Kb, Vb, bhBase, tid);

    for (int kt = 0; kt < NT; ++kt) {
        const int buf = kt & 1;
        if (kt + 1 < NT) {
            stage_kv(Ks[buf ^ 1], Vt[buf ^ 1], Kb, Vb,
                     bhBase + (size_t)(kt + 1) * 16 * D_MODEL, tid);
            WAIT_ASYNC(24);
        } else {
            WAIT_ASYNC(0);
        }
        __syncthreads();

        // ---- S = Q @ K^T: 32 WMMAs ----
        v8f sacc = {};
        const float* KsB = Ks[buf];
        #pragma unroll
        for (int kc = 0; kc < 32; ++kc) {
            v2f bf = *(const v2f*)(&KsB[r * KSTR + kc * 4 + 2 * half]);
            sacc = wmma_f32(qf[kc], bf, sacc);
        }

        // ---- scale + bias + softcap ----
        float sv[8];
        const int keyG = kt * 16 + r;
        #pragma unroll
        for (int p = 0; p < 8; ++p) {
            int qG  = qBase + p + 8 * half;
            float s = sacc[p] * scale + bias[(size_t)qG * SEQ + keyG];
            sv[p]   = tanhf(s * invCap) * cap;
        }

        // ---- row max over keys (16-lane half reduction) ----
        float tmax[8];
        #pragma unroll
        for (int p = 0; p < 8; ++p) tmax[p] = sv[p];
        #pragma unroll
        for (int m = 1; m < 16; m <<= 1)
            #pragma unroll
            for (int p = 0; p < 8; ++p)
                tmax[p] = fmaxf(tmax[p], __shfl_xor(tmax[p], m, 32));

        float c8[8], pv[8], tsum[8];
        #pragma unroll
        for (int p = 0; p < 8; ++p) {
            float mn = fmaxf(mrow[p], tmax[p]);
            c8[p]   = __expf(mrow[p] - mn);
            pv[p]   = __expf(sv[p] - mn);
            mrow[p] = mn;
            tsum[p] = pv[p];
        }
        #pragma unroll
        for (int m = 1; m < 16; m <<= 1)
            #pragma unroll
            for (int p = 0; p < 8; ++p)
                tsum[p] += __shfl_xor(tsum[p], m, 32);
        #pragma unroll
        for (int p = 0; p < 8; ++p) lrow[p] = lrow[p] * c8[p] + tsum[p];

        // ---- rescale running O ----
        #pragma unroll
        for (int nt = 0; nt < 8; ++nt)
            #pragma unroll
            for (int p = 0; p < 8; ++p)
                oacc[nt][p] *= c8[p];

        // ---- P: C-layout -> A-layout via per-wave LDS tile ----
        #pragma unroll
        for (int p = 0; p < 8; ++p)
            Pst[wave][(p + 8 * half) * PSTR + r] = pv[p];
        asm volatile("s_wait_dscnt 0" ::: "memory");

        // ---- O += P @ V: 32 WMMAs ----
        const float* VtB = Vt[buf];
        #pragma unroll
        for (int kc = 0; kc < 4; ++kc) {
            int k  = kc * 4 + 2 * half;
            v2f af = *(const v2f*)(&Pst[wave][r * PSTR + k]);
            #pragma unroll
            for (int nt = 0; nt < 8; ++nt) {
                v2f bf = *(const v2f*)(&VtB[(nt * 16 + r) * VSTR + k]);
                oacc[nt] = wmma_f32(af, bf, oacc[nt]);
            }
        }
        __syncthreads();
    }

    // ---- normalize, merge heads ----
    #pragma unroll
    for (int nt = 0; nt < 8; ++nt)
        #pragma unroll
        for (int p = 0; p < 8; ++p) {
            int qG = qBase + p + 8 * half;
            Ctx[(size_t)(b * SEQ + qG) * D_MODEL + headOff + nt * 16 + r] =
                oacc[nt][p] / lrow[p];
        }
}

// =============================================================================
// Host-side launch
// =============================================================================
extern "C" void kernel_launch(void* const* d_in, const int* in_sizes, int n_in,
                              void* d_out, int out_size, void* d_ws, size_t ws_size,
                              hipStream_t stream) {
    const float* inp  = (const float*)d_in[0];
    const float* wq   = (const float*)d_in[1];
    const float* wk   = (const float*)d_in[2];
    const float* wv   = (const float*)d_in[3];
    const float* wo   = (const float*)d_in[4];
    const float* bias = (const float*)d_in[5];
    const int*   cap  = (const int*)d_in[6];
    float* out = (float*)d_out;

    const int M = 2 * SEQ;                     // 4096 tokens
    const size_t mat = (size_t)M * D_MODEL;
    float* Qb  = (float*)d_ws;
    float* Kb  = Qb + mat;
    float* Vb  = Kb + mat;
    float* Ctx = Vb + mat;

    dim3 gproj(D_MODEL / BN, M / BM);          // (16, 32)
    gemm_xwt<<<gproj, dim3(256), 0, stream>>>(inp, wq, Qb, M, D_MODEL, D_MODEL);
    gemm_xwt<<<gproj, dim3(256), 0, stream>>>(inp, wk, Kb, M, D_MODEL, D_MODEL);
    gemm_xwt<<<gproj, dim3(256), 0, stream>>>(inp, wv, Vb, M, D_MODEL, D_MODEL);

    dim3 gattn(SEQ / 64, 2 * N_HEADS);         // (32, 32)
    attn_softcap<<<gattn, dim3(128), 0, stream>>>(Qb, Kb, Vb, bias, Ctx, cap);

    gemm_xwt<<<gproj, dim3(256), 0, stream>>>(Ctx, wo, out, M, D_MODEL, D_MODEL);
}